// MinkowskiSwitchNorm_35708358099270
// MI455X (gfx1250) — compile-verified
//
#include <hip/hip_runtime.h>

typedef __attribute__((ext_vector_type(2))) float v2f;
typedef __attribute__((ext_vector_type(4))) float v4f;
typedef __attribute__((ext_vector_type(8))) float v8f;

#define CHANNELS 64
#define NSEG 8
#define SN_EPS 1e-5f

// d_ws float layout:
// [0,512)      sum[8][64]
// [512,1024)   sumsq[8][64]
// [1024,1040)  counts (8 used)
// [1040,1552)  scale[8][64]
// [1552,2064)  shift[8][64]

__global__ void sn_init_ws(float* ws) {
    int i = blockIdx.x * blockDim.x + threadIdx.x;
    if (i < 1040) ws[i] = 0.0f;
}

__device__ __forceinline__ v8f sn_wmma(v2f a, v2f b, v8f c) {
    return __builtin_amdgcn_wmma_f32_16x16x4_f32(false, a, false, b, (short)0, c, false, false);
}

__global__ __launch_bounds__(128)
void sn_pass1_wmma(const float* __restrict__ x, const int* __restrict__ bids,
                   float* __restrict__ ws, int N) {
    const int lane = threadIdx.x & 31;
    const int wave = threadIdx.x >> 5;   // 0..3 -> channel group of 16
    const int m    = lane & 15;          // A: segment row ; B/D: channel col
    const int ksel = lane >> 4;          // 0 -> K={0,1}, 1 -> K={2,3}
    const int chan = wave * 16 + m;

    v8f acc_s = {};     // segment sum(x)
    v8f acc_q = {};     // segment sum(x^2)
    float cnt = 0.0f;   // per-lane count of (bid == m) over this lane's K slots

    const int stride = (int)gridDim.x * 4;
    int r = (int)blockIdx.x * 4;

    const float* xp = x + (long long)r * CHANNELS + 2 * ksel * CHANNELS + chan;
    const int*   bp = bids + r + 2 * ksel;
    const long long xstep = (long long)stride * CHANNELS;

    // Main loop: rows r..r+3 all valid -> unconditional loads, no exec masking.
    #pragma unroll 4
    for (; r + 3 < N; r += stride) {
        const int   b0 = bp[0];
        const int   b1 = bp[1];
        const float x0 = xp[0];
        const float x1 = xp[CHANNELS];
        v2f a;  a.x  = (b0 == m) ? 1.0f : 0.0f;  a.y  = (b1 == m) ? 1.0f : 0.0f;
        v2f bs; bs.x = x0;                        bs.y = x1;
        v2f bq; bq.x = x0 * x0;                   bq.y = x1 * x1;
        acc_s = sn_wmma(a, bs, acc_s);
        acc_q = sn_wmma(a, bq, acc_q);
        cnt += a.x + a.y;
        xp += xstep;
        bp += stride;
    }

    // Tail: at most one partial step (some of rows r..r+3 out of range).
    if (r < N) {
        const int r0 = r + 2 * ksel;
        const int r1 = r0 + 1;
        const bool ok0 = r0 < N, ok1 = r1 < N;
        const int b0 = ok0 ? bids[r0] : -1;
        const int b1 = ok1 ? bids[r1] : -1;
        const float x0 = ok0 ? x[(long long)r0 * CHANNELS + chan] : 0.0f;
        const float x1 = ok1 ? x[(long long)r1 * CHANNELS + chan] : 0.0f;
        v2f a;  a.x  = (b0 == m) ? 1.0f : 0.0f;  a.y  = (b1 == m) ? 1.0f : 0.0f;
        v2f bs; bs.x = x0;                        bs.y = x1;
        v2f bq; bq.x = x0 * x0;                   bq.y = x1 * x1;
        acc_s = sn_wmma(a, bs, acc_s);
        acc_q = sn_wmma(a, bq, acc_q);
        cnt += a.x + a.y;
    }

    // D layout: lanes 0..15 hold M=0..7 in VGPRs 0..7 (lanes 16..31 hold M=8..15, all zero here).
    if (lane < 16) {
        #pragma unroll
        for (int v = 0; v < 8; ++v) {
            atomicAdd(&ws[        v * CHANNELS + chan], acc_s[v]);
            atomicAdd(&ws[512   + v * CHANNELS + chan], acc_q[v]);
        }
    }
    // Counts: wave 0, all 32 lanes; lane m and lane m+16 both fold into counts[m].
    if (wave == 0) {
        atomicAdd(&ws[1024 + m], cnt);
    }
}

__global__ __launch_bounds__(64)
void sn_pass2_stats(float* __restrict__ ws,
                    const float* __restrict__ weight, const float* __restrict__ bias,
                    const float* __restrict__ mwv, const float* __restrict__ vwv, int N) {
    __shared__ float sm_mean[NSEG * CHANNELS];
    __shared__ float sm_ex2[NSEG * CHANNELS];
    __shared__ float s_mln[NSEG], s_vln[NSEG];

    const int c = threadIdx.x;  // 0..63
    float mean_in[NSEG], var_in[NSEG];
    float S = 0.0f, Q = 0.0f;
    for (int b = 0; b < NSEG; ++b) {
        const float cb = fmaxf(ws[1024 + b], 1.0f);
        const float s  = ws[       b * CHANNELS + c];
        const float q  = ws[512  + b * CHANNELS + c];
        S += s; Q += q;
        const float mi = s / cb;
        const float e2 = q / cb;
        mean_in[b] = mi;
        var_in[b]  = e2 - mi * mi;
        sm_mean[b * CHANNELS + c] = mi;
        sm_ex2[b * CHANNELS + c]  = e2;   // var_in + mean_in^2
    }
    __syncthreads();
    if (c < NSEG) {
        float a = 0.0f, e = 0.0f;
        for (int j = 0; j < CHANNELS; ++j) { a += sm_mean[c * CHANNELS + j]; e += sm_ex2[c * CHANNELS + j]; }
        a *= (1.0f / CHANNELS); e *= (1.0f / CHANNELS);
        s_mln[c] = a;
        s_vln[c] = e - a * a;
    }
    __syncthreads();

    const float fN = (float)N;
    const float mean_bn = S / fN;
    const float var_bn  = (Q - fN * mean_bn * mean_bn) / (fN - 1.0f);

    // softmax over the 3-vectors
    float m0 = mwv[0], m1 = mwv[1], m2 = mwv[2];
    float mx = fmaxf(m0, fmaxf(m1, m2));
    float e0 = expf(m0 - mx), e1 = expf(m1 - mx), e2 = expf(m2 - mx);
    float ei = 1.0f / (e0 + e1 + e2);
    const float mw0 = e0 * ei, mw1 = e1 * ei, mw2 = e2 * ei;
    float v0 = vwv[0], v1 = vwv[1], v2 = vwv[2];
    float vx = fmaxf(v0, fmaxf(v1, v2));
    float f0 = expf(v0 - vx), f1 = expf(v1 - vx), f2 = expf(v2 - vx);
    float fi = 1.0f / (f0 + f1 + f2);
    const float vw0 = f0 * fi, vw1 = f1 * fi, vw2 = f2 * fi;

    const float w = weight[c], bi = bias[c];
    for (int b = 0; b < NSEG; ++b) {
        const float mean = mw0 * mean_in[b] + mw1 * s_mln[b] + mw2 * mean_bn;
        const float var  = vw0 * var_in[b]  + vw1 * s_vln[b] + vw2 * var_bn;
        const float istd = 1.0f / sqrtf(var + SN_EPS);
        const float sc   = istd * w;
        ws[1040 + b * CHANNELS + c] = sc;
        ws[1552 + b * CHANNELS + c] = bi - mean * sc;
    }
}

__global__ __launch_bounds__(256)
void sn_pass3_apply(const float* __restrict__ x, const int* __restrict__ bids,
                    const float* __restrict__ ws, float* __restrict__ out, int N) {
    const long long total = (long long)N * (CHANNELS / 4);
    const long long gid = (long long)blockIdx.x * blockDim.x + threadIdx.x;
    if (gid >= total) return;
    const int row = (int)(gid >> 4);
    const int q   = (int)(gid & 15);
    const int b   = bids[row];
    const v4f xx = ((const v4f*)x)[gid];
    const v4f s  = ((const v4f*)(ws + 1040 + b * CHANNELS))[q];
    const v4f h  = ((const v4f*)(ws + 1552 + b * CHANNELS))[q];
    v4f o;
    o.x = fmaf(xx.x, s.x, h.x);
    o.y = fmaf(xx.y, s.y, h.y);
    o.z = fmaf(xx.z, s.z, h.z);
    o.w = fmaf(xx.w, s.w, h.w);
    __builtin_nontemporal_store(o, (v4f*)out + gid);
}

extern "C" void kernel_launch(void* const* d_in, const int* in_sizes, int n_in,
                              void* d_out, int out_size, void* d_ws, size_t ws_size,
                              hipStream_t stream) {
    const float* x      = (const float*)d_in[0];
    const int*   bids   = (const int*)d_in[1];
    const float* weight = (const float*)d_in[2];
    const float* bias   = (const float*)d_in[3];
    const float* mwv    = (const float*)d_in[4];
    const float* vwv    = (const float*)d_in[5];
    float* ws  = (float*)d_ws;
    float* out = (float*)d_out;
    const int N = in_sizes[0] / CHANNELS;

    sn_init_ws<<<(1040 + 255) / 256, 256, 0, stream>>>(ws);
    sn_pass1_wmma<<<2048, 128, 0, stream>>>(x, bids, ws, N);
    sn_pass2_stats<<<1, 64, 0, stream>>>(ws, weight, bias, mwv, vwv, N);
    const long long total = (long long)N * (CHANNELS / 4);
    const int grid3 = (int)((total + 255) / 256);
    sn_pass3_apply<<<grid3, 256, 0, stream>>>(x, bids, ws, out, N);
}